// GraphEncoder_62646392979690
// MI455X (gfx1250) — compile-verified
//
#include <hip/hip_runtime.h>
#include <hip/hip_bf16.h>

// ---------------------------------------------------------------------------
// GraphEncoder for MI455X (gfx1250, wave32).
//  - SpMM (Chebyshev propagation) : edge-parallel float4 gather + f32 atomic
//    scatter; h + accumulator fit in the 192MB L2, so this phase is L2-bound.
//  - Dense GEMMs                  : V_WMMA_F32_16X16X4_F32 (f32-exact WMMA),
//    compile-time CIN/COUT, LDS-staged A tile (conflict-free ds_load_b64).
//  - Final [16x320000]x[320000x128]: split-K WMMA, nontemporal W_last stream,
//    atomicAdd partials.
// ---------------------------------------------------------------------------

typedef float v2f __attribute__((ext_vector_type(2)));
typedef float v8f __attribute__((ext_vector_type(8)));

#define G_GROUPS 8
#define EPS_GN   1e-5f
#define FK_CHUNK 1280

__device__ __forceinline__ v8f wmma_f32_16x16x4(v2f a, v2f b, v8f c) {
  // 8 args: (neg_a, A, neg_b, B, c_mod, C, reuse_a, reuse_b)
  return __builtin_amdgcn_wmma_f32_16x16x4_f32(false, a, false, b, (short)0, c,
                                               false, false);
}

// ---------------------------------------------------------------------------
// Edge preprocessing
// ---------------------------------------------------------------------------

// int64 edge_index -> int32 row/col, accumulate degree (float counts).
__global__ void k_prep_edges(const long long* __restrict__ ei,
                             int* __restrict__ row32, int* __restrict__ col32,
                             float* __restrict__ deg, int E) {
  int e = blockIdx.x * blockDim.x + threadIdx.x;
  if (e >= E) return;
  int r = (int)ei[e];
  int c = (int)ei[(size_t)E + e];
  row32[e] = r;
  col32[e] = c;
  atomicAdd(&deg[r], 1.0f);
}

// deg -> deg^{-1/2} in place (deg==0 -> 0)
__global__ void k_deg_to_dis(float* __restrict__ deg, int Nn) {
  int i = blockIdx.x * blockDim.x + threadIdx.x;
  if (i >= Nn) return;
  float d = deg[i];
  deg[i] = (d > 0.f) ? rsqrtf(d) : 0.f;
}

// w[e] = -dis[row]*dis[col]
__global__ void k_edge_w(float* __restrict__ w, const float* __restrict__ dis,
                         const int* __restrict__ row32,
                         const int* __restrict__ col32, int E) {
  int e = blockIdx.x * blockDim.x + threadIdx.x;
  if (e >= E) return;
  w[e] = -dis[row32[e]] * dis[col32[e]];
}

// ---------------------------------------------------------------------------
// Propagation: dst[row] += scale * w[e] * src[col]   (accumulates into dst)
// ---------------------------------------------------------------------------

// cin divisible by 4: one wave-slice per edge (LANES = cin/4 float4 lanes).
template <int LANES>
__global__ void k_prop_vec4(float* __restrict__ dst,
                            const float* __restrict__ src,
                            const float* __restrict__ w,
                            const int* __restrict__ row32,
                            const int* __restrict__ col32, int E, float scale) {
  int e = blockIdx.x * blockDim.y + threadIdx.y;
  if (e >= E) return;
  int tx = threadIdx.x;  // 0..LANES-1
  float we = w[e] * scale;
  int cs = col32[e];
  int rs = row32[e];
  const float4 v = ((const float4*)(src + (size_t)cs * (LANES * 4)))[tx];
  float* d = dst + (size_t)rs * (LANES * 4) + tx * 4;
  atomicAdd(d + 0, we * v.x);
  atomicAdd(d + 1, we * v.y);
  atomicAdd(d + 2, we * v.z);
  atomicAdd(d + 3, we * v.w);
}

// cin == 3 (first layer): one thread per edge.
__global__ void k_prop3(float* __restrict__ dst, const float* __restrict__ src,
                        const float* __restrict__ w,
                        const int* __restrict__ row32,
                        const int* __restrict__ col32, int E, float scale) {
  int e = blockIdx.x * blockDim.x + threadIdx.x;
  if (e >= E) return;
  float we = w[e] * scale;
  int cs = col32[e];
  int rs = row32[e];
  atomicAdd(dst + (size_t)rs * 3 + 0, we * src[(size_t)cs * 3 + 0]);
  atomicAdd(dst + (size_t)rs * 3 + 1, we * src[(size_t)cs * 3 + 1]);
  atomicAdd(dst + (size_t)rs * 3 + 2, we * src[(size_t)cs * 3 + 2]);
}

// dst[i] = -src[i]   (init for Tx_k = 2*prop(Tx_{k-1}) - Tx_{k-2})
__global__ void k_negcopy4(float4* __restrict__ dst,
                           const float4* __restrict__ src, int n4) {
  int i = blockIdx.x * blockDim.x + threadIdx.x;
  if (i >= n4) return;
  float4 v = src[i];
  dst[i] = make_float4(-v.x, -v.y, -v.z, -v.w);
}

// ---------------------------------------------------------------------------
// GEMM accumulate: out[N,COUT] += A[N,CIN] @ Wk[CIN,COUT]
// Compile-time CIN/COUT. One wave = one 16x16 output tile; the shared 16xCIN
// A tile is staged once per block into padded LDS (stride CIN+4 floats ->
// conflict-free ds_load_b64: lanes hit banks 0,4,...,60).
// block = 32*(COUT/16) threads; grid = N/16.
// ---------------------------------------------------------------------------
template <int CIN, int COUT>
__global__ void k_gemm_accum(float* __restrict__ out,
                             const float* __restrict__ A,
                             const float* __restrict__ Wk) {
  constexpr int STRIDE = CIN + 4;        // float padding vs 64 banks
  constexpr int NWAVE = COUT / 16;
  __shared__ float As[16 * STRIDE];

  int tid = threadIdx.x;
  int m0 = blockIdx.x * 16;

  // Cooperative A-tile load (16 x CIN floats) at float4 granularity.
  constexpr int NF4 = 16 * CIN / 4;
#pragma unroll
  for (int i = tid; i < NF4; i += 32 * NWAVE) {
    int r = i / (CIN / 4);
    int cq = i % (CIN / 4);
    float4 v = ((const float4*)(A + (size_t)(m0 + r) * CIN))[cq];
    *(float4*)(&As[r * STRIDE + cq * 4]) = v;
  }
  __syncthreads();

  int lane = tid & 31;
  int wave = tid >> 5;
  int n0 = wave * 16;
  int idx = lane & 15;              // A: row m; B/C/D: column n
  int khalf = (lane >> 4) << 1;     // 0 or 2
  int mbase = (lane < 16) ? 0 : 8;  // C/D row base

  float* crow = out + (size_t)(m0 + mbase) * COUT + n0 + idx;
  v8f c;
#pragma unroll
  for (int v = 0; v < 8; ++v) c[v] = crow[v * COUT];

  const float* Als = &As[idx * STRIDE + khalf];
  const float* Bcol = Wk + (size_t)khalf * COUT + n0 + idx;
#pragma unroll 8
  for (int ks = 0; ks < CIN / 4; ++ks) {
    v2f a = *(const v2f*)(Als + ks * 4);  // ds_load_b64, conflict-free
    v2f b;
    b.x = Bcol[(ks * 4) * COUT];
    b.y = Bcol[(ks * 4 + 1) * COUT];
    c = wmma_f32_16x16x4(a, b, c);
  }

#pragma unroll
  for (int v = 0; v < 8; ++v) crow[v * COUT] = c[v];
}

// CIN == 3 special case (layer 0): single WMMA step. The K=3 tail is handled
// with in-bounds loads times a lane mask (v_cndmask) -- no divergent loads.
template <int COUT>
__global__ void k_gemm_c3(float* __restrict__ out, const float* __restrict__ A,
                          const float* __restrict__ Wk) {
  int lane = threadIdx.x & 31;
  int wave = threadIdx.x >> 5;
  int m0 = blockIdx.x * 16;
  int n0 = wave * 16;
  int idx = lane & 15;
  int khalf = (lane >> 4) << 1;
  int mbase = (lane < 16) ? 0 : 8;
  float mk = (khalf == 0) ? 1.f : 0.f;  // zero out the k=3 slot

  const float* Arow = A + (size_t)(m0 + idx) * 3;
  v2f a, b;
  a.x = Arow[khalf];        // k = 0 or 2 (always < 3)
  a.y = Arow[1] * mk;       // k = 1, or masked 0 for upper half
  b.x = Wk[khalf * COUT + n0 + idx];
  b.y = Wk[COUT + n0 + idx] * mk;

  float* crow = out + (size_t)(m0 + mbase) * COUT + n0 + idx;
  v8f c;
#pragma unroll
  for (int v = 0; v < 8; ++v) c[v] = crow[v * COUT];
  c = wmma_f32_16x16x4(a, b, c);
#pragma unroll
  for (int v = 0; v < 8; ++v) crow[v * COUT] = c[v];
}

// ---------------------------------------------------------------------------
// GroupNorm(+bias) + ReLU, in place, float4-vectorized.
// One thread per (node, group); a wave touches 32*CG*4B contiguous bytes.
// ---------------------------------------------------------------------------
template <int CG>
__global__ void k_gn_relu(float* __restrict__ h, const float* __restrict__ bias,
                          const float* __restrict__ gamma,
                          const float* __restrict__ beta, int Nn) {
  int t = blockIdx.x * blockDim.x + threadIdx.x;
  if (t >= Nn * G_GROUPS) return;
  int g = t & (G_GROUPS - 1);
  int n = t >> 3;
  const int C = CG * G_GROUPS;
  constexpr int V = CG / 4;

  float4* p4 = (float4*)(h + (size_t)n * C + g * CG);
  const float4* b4 = (const float4*)(bias + g * CG);
  const float4* g4 = (const float4*)(gamma + g * CG);
  const float4* t4 = (const float4*)(beta + g * CG);

  float4 xs[V];
  float s = 0.f, s2 = 0.f;
#pragma unroll
  for (int i = 0; i < V; ++i) {
    float4 v = p4[i];
    float4 bb = b4[i];
    v.x += bb.x; v.y += bb.y; v.z += bb.z; v.w += bb.w;
    xs[i] = v;
    s += v.x + v.y + v.z + v.w;
    s2 += v.x * v.x + v.y * v.y + v.z * v.z + v.w * v.w;
  }
  float mu = s * (1.f / CG);
  float inv = rsqrtf(s2 * (1.f / CG) - mu * mu + EPS_GN);
#pragma unroll
  for (int i = 0; i < V; ++i) {
    float4 v = xs[i], gg = g4[i], bt = t4[i], o;
    o.x = fmaxf((v.x - mu) * inv * gg.x + bt.x, 0.f);
    o.y = fmaxf((v.y - mu) * inv * gg.y + bt.y, 0.f);
    o.z = fmaxf((v.z - mu) * inv * gg.z + bt.z, 0.f);
    o.w = fmaxf((v.w - mu) * inv * gg.w + bt.w, 0.f);
    p4[i] = o;
  }
}

// ---------------------------------------------------------------------------
// Final head
// ---------------------------------------------------------------------------
__global__ void k_bias_init(float* __restrict__ out,
                            const float* __restrict__ b_last, int total) {
  int i = blockIdx.x * blockDim.x + threadIdx.x;
  if (i < total) out[i] = b_last[i & 127];
}

// out[16,128] += A[16,Ktot] @ B[Ktot,128], split-K. grid = K chunks,
// block = 8 waves (one 16x16 n-tile each). W_last is streamed exactly once ->
// nontemporal loads keep it from churning the L2. Partials via f32 atomics.
__global__ void k_final_gemm(float* __restrict__ out,
                             const float* __restrict__ A,
                             const float* __restrict__ B, int Ktot) {
  int lane = threadIdx.x & 31;
  int wave = threadIdx.x >> 5;  // 0..7 -> n tile
  int n0 = wave * 16;
  int col = lane & 15;
  int khalf = (lane >> 4) << 1;
  int kstart = blockIdx.x * FK_CHUNK;
  int kend = kstart + FK_CHUNK;
  if (kend > Ktot) kend = Ktot;

  v8f c = {0.f, 0.f, 0.f, 0.f, 0.f, 0.f, 0.f, 0.f};
  const float* Arow = A + (size_t)col * Ktot + khalf;  // batch row (contig.)
  const float* Bc = B + n0 + col;
#pragma unroll 4
  for (int k0 = kstart; k0 < kend; k0 += 4) {
    v2f a = *(const v2f*)(Arow + k0);  // aligned b64
    v2f b;
    b.x = __builtin_nontemporal_load(Bc + (size_t)(k0 + khalf) * 128);
    b.y = __builtin_nontemporal_load(Bc + (size_t)(k0 + khalf + 1) * 128);
    c = wmma_f32_16x16x4(a, b, c);
  }
  int mbase = (lane < 16) ? 0 : 8;
#pragma unroll
  for (int v = 0; v < 8; ++v)
    atomicAdd(&out[(size_t)(mbase + v) * 128 + n0 + col], c[v]);
}

// ---------------------------------------------------------------------------
// Host orchestration
// ---------------------------------------------------------------------------
static void launch_prop(float* dst, const float* src, const float* w,
                        const int* row32, const int* col32, int E, float scale,
                        int cin, hipStream_t s) {
  if (cin == 3) {
    k_prop3<<<(E + 255) / 256, 256, 0, s>>>(dst, src, w, row32, col32, E, scale);
  } else if (cin == 128) {
    dim3 blk(32, 8);
    k_prop_vec4<32><<<(E + 7) / 8, blk, 0, s>>>(dst, src, w, row32, col32, E, scale);
  } else {  // cin == 64
    dim3 blk(16, 16);
    k_prop_vec4<16><<<(E + 15) / 16, blk, 0, s>>>(dst, src, w, row32, col32, E, scale);
  }
}

static void launch_gemm(float* out, const float* A, const float* Wk, int layer,
                        int Nn, hipStream_t s) {
  if (layer == 0)
    k_gemm_c3<128><<<Nn / 16, 256, 0, s>>>(out, A, Wk);
  else if (layer == 1)
    k_gemm_accum<128, 64><<<Nn / 16, 128, 0, s>>>(out, A, Wk);
  else
    k_gemm_accum<64, 32><<<Nn / 16, 64, 0, s>>>(out, A, Wk);
}

static void launch_negcopy(float* dst, const float* src, size_t nflt,
                           hipStream_t s) {
  size_t n4 = nflt / 4;
  k_negcopy4<<<(unsigned)((n4 + 255) / 256), 256, 0, s>>>(
      (float4*)dst, (const float4*)src, (int)n4);
}

extern "C" void kernel_launch(void* const* d_in, const int* in_sizes, int n_in,
                              void* d_out, int out_size, void* d_ws,
                              size_t ws_size, hipStream_t stream) {
  const float* x = (const float*)d_in[0];           // [N,3]
  const long long* ei = (const long long*)d_in[1];  // [2,E] int64
  const float* Wl[3] = {(const float*)d_in[2], (const float*)d_in[6],
                        (const float*)d_in[10]};
  const float* bl[3] = {(const float*)d_in[3], (const float*)d_in[7],
                        (const float*)d_in[11]};
  const float* gl[3] = {(const float*)d_in[4], (const float*)d_in[8],
                        (const float*)d_in[12]};
  const float* be[3] = {(const float*)d_in[5], (const float*)d_in[9],
                        (const float*)d_in[13]};
  const float* W_last = (const float*)d_in[14];  // [N*2, 128]
  const float* b_last = (const float*)d_in[15];  // [128]

  const int N = in_sizes[0] / 3;  // 160000
  const int E = in_sizes[1] / 2;  // 2560000
  const int cins[3] = {3, 128, 64};
  const int couts[3] = {128, 64, 32};

  // Workspace carve-up (256B aligned slabs)
  char* p = (char*)d_ws;
  auto carve = [&](size_t bytes) -> void* {
    void* r = (void*)p;
    p += (bytes + 255) & ~(size_t)255;
    return r;
  };
  float* deg = (float*)carve((size_t)N * 4);  // degree -> dis (in place)
  int* row32 = (int*)carve((size_t)E * 4);
  int* col32 = (int*)carve((size_t)E * 4);
  float* w = (float*)carve((size_t)E * 4);
  size_t big = (size_t)N * 128 * 4;
  float* T0 = (float*)carve(big);
  float* T1 = (float*)carve(big);
  float* T2 = (float*)carve(big);
  float* H0 = (float*)carve(big);
  float* H1 = (float*)carve(big);

  // --- Edge preprocessing ---------------------------------------------------
  hipMemsetAsync(deg, 0, (size_t)N * 4, stream);
  k_prep_edges<<<(E + 255) / 256, 256, 0, stream>>>(ei, row32, col32, deg, E);
  k_deg_to_dis<<<(N + 255) / 256, 256, 0, stream>>>(deg, N);
  k_edge_w<<<(E + 255) / 256, 256, 0, stream>>>(w, deg, row32, col32, E);

  // --- Three ChebConv + GroupNorm + ReLU layers -----------------------------
  const float* hin = x;
  for (int l = 0; l < 3; ++l) {
    int cin = cins[l], cout = couts[l];
    float* hout = (l == 1) ? H0 : H1;  // l0->H1, l1->H0, l2->H1
    size_t nflt = (size_t)N * cin;
    const float* W = Wl[l];
    size_t wk = (size_t)cin * cout;

    hipMemsetAsync(hout, 0, (size_t)N * cout * 4, stream);
    // k = 0 : Tx0 = hin
    launch_gemm(hout, hin, W + 0 * wk, l, N, stream);
    // k = 1 : Tx1 = prop(Tx0)
    hipMemsetAsync(T0, 0, nflt * 4, stream);
    launch_prop(T0, hin, w, row32, col32, E, 1.0f, cin, stream);
    launch_gemm(hout, T0, W + 1 * wk, l, N, stream);
    // k = 2 : Tx2 = 2*prop(Tx1) - Tx0
    launch_negcopy(T1, hin, nflt, stream);
    launch_prop(T1, T0, w, row32, col32, E, 2.0f, cin, stream);
    launch_gemm(hout, T1, W + 2 * wk, l, N, stream);
    // k = 3 : Tx3 = 2*prop(Tx2) - Tx1
    launch_negcopy(T2, T0, nflt, stream);
    launch_prop(T2, T1, w, row32, col32, E, 2.0f, cin, stream);
    launch_gemm(hout, T2, W + 3 * wk, l, N, stream);
    // k = 4 : Tx4 = 2*prop(Tx3) - Tx2
    launch_negcopy(T0, T1, nflt, stream);
    launch_prop(T0, T2, w, row32, col32, E, 2.0f, cin, stream);
    launch_gemm(hout, T0, W + 4 * wk, l, N, stream);
    // k = 5 : Tx5 = 2*prop(Tx4) - Tx3
    launch_negcopy(T1, T2, nflt, stream);
    launch_prop(T1, T0, w, row32, col32, E, 2.0f, cin, stream);
    launch_gemm(hout, T1, W + 5 * wk, l, N, stream);

    // GroupNorm(+bias) + ReLU, in place
    int total = N * G_GROUPS;
    if (cout == 128)
      k_gn_relu<16><<<(total + 255) / 256, 256, 0, stream>>>(hout, bl[l], gl[l],
                                                             be[l], N);
    else if (cout == 64)
      k_gn_relu<8><<<(total + 255) / 256, 256, 0, stream>>>(hout, bl[l], gl[l],
                                                            be[l], N);
    else
      k_gn_relu<4><<<(total + 255) / 256, 256, 0, stream>>>(hout, bl[l], gl[l],
                                                            be[l], N);
    hin = hout;
  }

  // --- Final dense head: [16, N*2] @ [N*2, 128] -----------------------------
  // Row-major reshape of h3 [N,32] -> [16, V_CNT*32] is a no-op: contiguous.
  int Ktot = N * 2;  // V_CNT * 32
  float* out = (float*)d_out;
  k_bias_init<<<(16 * 128 + 255) / 256, 256, 0, stream>>>(out, b_last,
                                                          16 * 128);
  int nchunks = (Ktot + FK_CHUNK - 1) / FK_CHUNK;
  k_final_gemm<<<nchunks, 256, 0, stream>>>(out, hin, W_last, Ktot);
}